// BahdanauAttention_74586402063274
// MI455X (gfx1250) — compile-verified
//
#include <hip/hip_runtime.h>

#define H 512
#define HP 516                  // padded LDS row stride (16B-aligned, bank-spread)
#define BATCH 32
#define SEQ 4096
#define CHUNK 128
#define NCHUNK (SEQ / CHUNK)    // 32 chunks per batch

typedef __attribute__((ext_vector_type(2))) float v2f;
typedef __attribute__((ext_vector_type(8))) float v8f;

// ---------------------------------------------------------------------------
// Kernel A: combo[b,k] = decoder_hidden[b,:] @ W_dec[:,k] + b_dec[k] + b_enc[k]
// ---------------------------------------------------------------------------
__global__ __launch_bounds__(256) void dec_proj_kernel(
    const float* __restrict__ dh, const float* __restrict__ Wd,
    const float* __restrict__ bd, const float* __restrict__ be,
    float* __restrict__ combo) {
  __shared__ float sdh[H];
  int b = blockIdx.x;
  for (int i = threadIdx.x; i < H; i += blockDim.x) sdh[i] = dh[b * H + i];
  __syncthreads();
  for (int k = threadIdx.x; k < H; k += blockDim.x) {
    float acc = bd[k] + be[k];
    for (int j = 0; j < H; ++j) acc = fmaf(sdh[j], Wd[j * H + k], acc);
    combo[b * H + k] = acc;
  }
}

// Per-thread share of one 128x32 W_enc slab (4 x float4 = 16 VGPRs)
struct SlabRegs {
  float4 a0, a1, b0, b1;
};

__device__ __forceinline__ SlabRegs load_slab(const float* __restrict__ We,
                                              int K0, int n0, int stKp0,
                                              int stN4) {
  SlabRegs r;
  r.a0 = *(const float4*)(We + (size_t)(K0 + 2 * stKp0) * H + n0 + stN4);
  r.a1 = *(const float4*)(We + (size_t)(K0 + 2 * stKp0 + 1) * H + n0 + stN4);
  r.b0 = *(const float4*)(We + (size_t)(K0 + 2 * (stKp0 + 32)) * H + n0 + stN4);
  r.b1 = *(const float4*)(We + (size_t)(K0 + 2 * (stKp0 + 32) + 1) * H + n0 + stN4);
  return r;
}

__device__ __forceinline__ void store_slab(float* __restrict__ buf,
                                           const SlabRegs& r, int stKp0,
                                           int stN4) {
  // K-pair interleave: buf[(kp*32+n)*2 + {0,1}] = row 2kp, row 2kp+1 at col n
  float* dA = buf + (stKp0 * 32 + stN4) * 2;
  float* dB = buf + ((stKp0 + 32) * 32 + stN4) * 2;
  *(float4*)(dA)     = make_float4(r.a0.x, r.a1.x, r.a0.y, r.a1.y);
  *(float4*)(dA + 4) = make_float4(r.a0.z, r.a1.z, r.a0.w, r.a1.w);
  *(float4*)(dB)     = make_float4(r.b0.x, r.b1.x, r.b0.y, r.b1.y);
  *(float4*)(dB + 4) = make_float4(r.b0.z, r.b1.z, r.b0.w, r.b1.w);
}

// ---------------------------------------------------------------------------
// Kernel B: fused  enc@W_enc -> tanh -> score -> chunk softmax partials
//           + partial context (single pass over encoder_outputs)
// grid = (NCHUNK, BATCH), block = 256 threads = 8 wave32
// Each wave owns one 16-row M-tile; per staged 32-col W_enc slab it computes
// both 16-col N-tiles from one A fragment. W_enc slabs are double-buffered in
// LDS and loaded via a register-staged software pipeline (loads for slab i+1
// issue before the 64-WMMA compute block on slab i).
// ---------------------------------------------------------------------------
__global__ __launch_bounds__(256) void attn_main_kernel(
    const float* __restrict__ enc, const float* __restrict__ We,
    const float* __restrict__ combo, const float* __restrict__ Wv,
    const float* __restrict__ bv,
    float* __restrict__ scores, float* __restrict__ wsM,
    float* __restrict__ wsL, float* __restrict__ wsCtx) {
  __shared__ float ldsEnc[CHUNK * HP];  // ~258 KB: enc chunk (padded rows)
  __shared__ float ldsB[2][128 * 32];   // 2 x 16 KB: double-buffered W_enc slab
  __shared__ float ldsCombo[H];
  __shared__ float ldsWv[H];
  __shared__ float sPart[8][16];        // per-wave per-row score partials
  __shared__ float sScore[CHUNK];
  __shared__ float sEw[CHUNK];
  __shared__ float sMax;

  const int b = blockIdx.y;
  const int chunk = blockIdx.x;
  const int s0 = chunk * CHUNK;
  const int tid = threadIdx.x;
  const int wave = tid >> 5;            // 8 waves (wave32)
  const int lane = tid & 31;

  // Stage enc chunk into LDS with padded row stride (bank-conflict-free A reads)
  const float* encBase = enc + ((size_t)b * SEQ + s0) * H;
  for (int i = tid; i < CHUNK * (H / 4); i += 256) {
    int r = i >> 7;                     // H/4 = 128 float4 per row
    int c = (i & 127) * 4;
    *(float4*)(ldsEnc + r * HP + c) = *(const float4*)(encBase + r * H + c);
  }
  for (int i = tid; i < H; i += 256) {
    ldsCombo[i] = combo[b * H + i];
    ldsWv[i] = Wv[i];
  }

  const int mRow = wave * 16;           // wave's M-tile
  const int aRow = mRow + (lane & 15);
  const int kHalf = lane >> 4;          // 0 or 1 : lane-half K offset/2
  const int col = lane & 15;

  // B staging unit mapping: kpair stKp0 (+32), 4 consecutive n columns
  const int stN4 = (tid & 7) * 4;
  const int stKp0 = tid >> 3;

  float rAcc[8];
#pragma unroll
  for (int v = 0; v < 8; ++v) rAcc[v] = 0.0f;

  // Pipeline prologue: slab 0 (np=0, ks=0)
  SlabRegs regs = load_slab(We, 0, 0, stKp0, stN4);
  store_slab(ldsB[0], regs, stKp0, stN4);
  __syncthreads();

  for (int np = 0; np < 16; ++np) {
    const int n0 = np * 32;
    v8f acc0 = {};
    v8f acc1 = {};
    for (int ks = 0; ks < 4; ++ks) {
      const int K0 = ks * 128;
      const int it = np * 4 + ks;
      const int cur = it & 1;
      const bool hasNext = (it + 1) < 64;
      // issue global loads for slab it+1 now; consume after compute
      if (hasNext) {
        const int nit = it + 1;
        regs = load_slab(We, (nit & 3) * 128, (nit >> 2) * 32, stKp0, stN4);
      }
      // warm L2/L0 two slabs ahead
      if (it + 2 < 64) {
        const int pit = it + 2;
        const size_t ro =
            (size_t)((pit & 3) * 128 + 2 * stKp0) * H + (pit >> 2) * 32 + stN4;
        __builtin_prefetch(We + ro, 0, 3);
        __builtin_prefetch(We + ro + H, 0, 3);
        __builtin_prefetch(We + ro + (size_t)64 * H, 0, 3);
        __builtin_prefetch(We + ro + (size_t)65 * H, 0, 3);
      }
      const float* Bc = ldsB[cur];
#pragma unroll 4
      for (int ki = 0; ki < 32; ++ki) {
        const int k0 = ki * 4;
        // A frag (16x4 f32): lane=M row, VGPRs = K{2*kHalf, 2*kHalf+1}
        v2f a = *(const v2f*)(ldsEnc + aRow * HP + K0 + k0 + 2 * kHalf);
        // B frags: ds_load_2addr_b64 pair thanks to K-pair interleave
        const int kpair = 2 * ki + kHalf;
        v2f bb0 = *(const v2f*)(Bc + (kpair * 32 + col) * 2);
        v2f bb1 = *(const v2f*)(Bc + (kpair * 32 + 16 + col) * 2);
        acc0 = __builtin_amdgcn_wmma_f32_16x16x4_f32(
            false, a, false, bb0, (short)0, acc0, false, false);
        acc1 = __builtin_amdgcn_wmma_f32_16x16x4_f32(
            false, a, false, bb1, (short)0, acc1, false, false);
      }
      if (hasNext) store_slab(ldsB[cur ^ 1], regs, stKp0, stN4);
      __syncthreads();
    }
    // Epilogue both tiles: element (M,N) at vgpr v, lane l:
    //   M = mRow + v + 8*(l>>4),  N = n0 + nh*16 + (l&15)
#pragma unroll
    for (int nh = 0; nh < 2; ++nh) {
      const int ncol = n0 + nh * 16 + col;
      const float wv = ldsWv[ncol];
      const float cb = ldsCombo[ncol];
#pragma unroll
      for (int v = 0; v < 8; ++v) {
        float t = tanhf((nh ? acc1[v] : acc0[v]) + cb) * wv;
        t += __shfl_xor(t, 1, 32);
        t += __shfl_xor(t, 2, 32);
        t += __shfl_xor(t, 4, 32);
        t += __shfl_xor(t, 8, 32);
        rAcc[v] += t;                   // valid on lanes 0 and 16
      }
    }
  }

  // lanes 0/16 hold per-row score partials for rows mRow+v / mRow+8+v
  if (col == 0) {
    const int base = kHalf * 8;
#pragma unroll
    for (int v = 0; v < 8; ++v) sPart[wave][base + v] = rAcc[v];
  }
  __syncthreads();

  // add b_v, emit raw scores
  if (tid < CHUNK) {
    float s = sPart[tid >> 4][tid & 15] + bv[0];
    sScore[tid] = s;
    scores[(size_t)b * SEQ + s0 + tid] = s;
  }
  __syncthreads();
  if (tid == 0) {
    float m = -3.402823466e+38f;
    for (int r = 0; r < CHUNK; ++r) m = fmaxf(m, sScore[r]);
    sMax = m;
  }
  __syncthreads();
  if (tid < CHUNK) sEw[tid] = expf(sScore[tid] - sMax);
  __syncthreads();
  if (tid == 0) {
    float l = 0.0f;
    for (int r = 0; r < CHUNK; ++r) l += sEw[r];
    const int ci = b * NCHUNK + chunk;
    wsM[ci] = sMax;
    wsL[ci] = l;
  }
  __syncthreads();
  // partial context from the LDS-resident enc chunk (no second HBM pass)
  for (int h = tid; h < H; h += 256) {
    float c = 0.0f;
    for (int r = 0; r < CHUNK; ++r) c = fmaf(sEw[r], ldsEnc[r * HP + h], c);
    wsCtx[((size_t)b * NCHUNK + chunk) * H + h] = c;
  }
}

// ---------------------------------------------------------------------------
// Kernel C: cross-chunk softmax combine -> context + attention weights
// ---------------------------------------------------------------------------
__global__ __launch_bounds__(256) void finalize_kernel(
    const float* __restrict__ scores, const float* __restrict__ wsM,
    const float* __restrict__ wsL, const float* __restrict__ wsCtx,
    float* __restrict__ outCtx, float* __restrict__ outW) {
  __shared__ float sScale[NCHUNK];
  __shared__ float sMg, sLg;
  const int b = blockIdx.x;
  const int tid = threadIdx.x;
  if (tid == 0) {
    float m = -3.402823466e+38f;
    for (int c = 0; c < NCHUNK; ++c) m = fmaxf(m, wsM[b * NCHUNK + c]);
    float l = 0.0f;
    for (int c = 0; c < NCHUNK; ++c) {
      float sc = expf(wsM[b * NCHUNK + c] - m);
      sScale[c] = sc;
      l += wsL[b * NCHUNK + c] * sc;
    }
    sMg = m;
    sLg = l;
  }
  __syncthreads();
  const float m = sMg;
  const float invL = 1.0f / sLg;
  for (int s = tid; s < SEQ; s += blockDim.x) {
    outW[(size_t)b * SEQ + s] = expf(scores[(size_t)b * SEQ + s] - m) * invL;
  }
  for (int h = tid; h < H; h += blockDim.x) {
    float c = 0.0f;
    for (int cc = 0; cc < NCHUNK; ++cc) {
      c += wsCtx[((size_t)b * NCHUNK + cc) * H + h] * sScale[cc];
    }
    outCtx[(size_t)b * H + h] = c * invL;
  }
}

// ---------------------------------------------------------------------------
extern "C" void kernel_launch(void* const* d_in, const int* in_sizes, int n_in,
                              void* d_out, int out_size, void* d_ws,
                              size_t ws_size, hipStream_t stream) {
  const float* enc = (const float*)d_in[0];  // (B, S, H)
  const float* dh  = (const float*)d_in[1];  // (B, H)
  const float* We  = (const float*)d_in[2];  // (H, H)
  const float* be  = (const float*)d_in[3];  // (H,)
  const float* Wd  = (const float*)d_in[4];  // (H, H)
  const float* bd  = (const float*)d_in[5];  // (H,)
  const float* Wv  = (const float*)d_in[6];  // (H, 1)
  const float* bv  = (const float*)d_in[7];  // (1,)

  float* out = (float*)d_out;
  float* outCtx = out;                 // (B, H) first (tuple order)
  float* outW = out + BATCH * H;       // (B, S) second

  float* ws = (float*)d_ws;
  float* combo  = ws;                               // B*H
  float* scores = combo + BATCH * H;                // B*S
  float* wsM    = scores + (size_t)BATCH * SEQ;     // B*NCHUNK
  float* wsL    = wsM + BATCH * NCHUNK;             // B*NCHUNK
  float* wsCtx  = wsL + BATCH * NCHUNK;             // B*NCHUNK*H

  dec_proj_kernel<<<dim3(BATCH), dim3(256), 0, stream>>>(dh, Wd, bd, be, combo);
  attn_main_kernel<<<dim3(NCHUNK, BATCH), dim3(256), 0, stream>>>(
      enc, We, combo, Wv, bv, scores, wsM, wsL, wsCtx);
  finalize_kernel<<<dim3(BATCH), dim3(256), 0, stream>>>(
      scores, wsM, wsL, wsCtx, outCtx, outW);
}